// WHDRHingeLossParaModule_45423574122779
// MI455X (gfx1250) — compile-verified
//
#include <hip/hip_runtime.h>
#include <cstdint>

// ---------------------------------------------------------------------------
// WHDR hinge loss for MI455X (gfx1250).
// Memory-bound streaming reduction: 96 MB target stream (HBM, ~4.2us floor at
// 23.3 TB/s) + random 4B gathers into a 4 MB L2-resident image. No matrix math
// -> no WMMA; instead we use the CDNA5 async-tensor data path:
// GLOBAL_LOAD_ASYNC_TO_LDS_B128 double-buffered tiles tracked with ASYNCcnt
// (s_wait_asynccnt), compute from LDS, wave32 shuffle reduction, per-block
// partials in d_ws, then a tiny finalize kernel computes the division.
// ---------------------------------------------------------------------------

#define TILE_PAIRS 1024                       // pairs per LDS tile
#define TILE_BYTES (TILE_PAIRS * 24)          // 24 KB per tile
#define NTHREADS   256                        // 8 waves (wave32)
#define B128S      (TILE_BYTES / (NTHREADS * 16))  // 6 async B128 per thread
#define PPT        (TILE_PAIRS / NTHREADS)         // 4 pairs per thread
#define NBLOCKS    1024

typedef int v4i __attribute__((ext_vector_type(4)));
typedef __attribute__((address_space(1))) v4i* global_v4i_ptr;
typedef __attribute__((address_space(3))) v4i* lds_v4i_ptr;

__device__ __forceinline__ void async_copy_b128(const void* gsrc, void* lds_dst) {
#if defined(__AMDGCN__)
#if __has_builtin(__builtin_amdgcn_global_load_async_to_lds_b128)
  __builtin_amdgcn_global_load_async_to_lds_b128(
      (global_v4i_ptr)gsrc, (lds_v4i_ptr)lds_dst, /*offset=*/0, /*cpol=*/0);
#else
  uint32_t loff = (uint32_t)(uintptr_t)lds_dst;  // flat LDS addr truncates to DS offset
  asm volatile("global_load_async_to_lds_b128 %0, %1, off"
               :: "v"(loff), "v"(gsrc) : "memory");
#endif
#endif
}

template <int N>
__device__ __forceinline__ void wait_async() {
#if defined(__AMDGCN__)
#if __has_builtin(__builtin_amdgcn_s_wait_asynccnt)
  __builtin_amdgcn_s_wait_asynccnt(N);
#else
  asm volatile("s_wait_asynccnt %0" :: "n"(N) : "memory");
#endif
#endif
}

__global__ __launch_bounds__(NTHREADS) void whdr_reduce_kernel(
    const float* __restrict__ img, const float* __restrict__ target,
    float* __restrict__ partials, int nPairs, int nTiles) {
  __shared__ float4 buf[2][TILE_BYTES / 16];     // 2 x 24 KB double buffer
  __shared__ float wavePart[NTHREADS / 32][2];

  const int tid = threadIdx.x;
  const int bid = blockIdx.x;
  const int gsz = gridDim.x;
  const size_t totalBytes = (size_t)nPairs * 24u;

  // tiles handled by this block: bid, bid+gsz, ... (uniform count per block)
  const int count = (nTiles > bid) ? ((nTiles - bid + gsz - 1) / gsz) : 0;

  float accWM = 0.f, accW = 0.f;

  if (count > 0) {  // prologue: issue tile 0 into buffer 0
    const size_t base = (size_t)bid * TILE_BYTES;
#pragma unroll
    for (int k = 0; k < B128S; ++k) {
      const size_t off = base + (size_t)(tid * 16 + k * (NTHREADS * 16));
      if (off + 16 <= totalBytes)
        async_copy_b128((const char*)target + off,
                        (char*)buf[0] + tid * 16 + k * (NTHREADS * 16));
    }
  }

  for (int i = 0; i < count; ++i) {
    const int tile = bid + i * gsz;
    if (i + 1 < count) {  // issue next tile, then wait for current one (<=B128S outstanding)
      const size_t base = (size_t)(bid + (i + 1) * gsz) * TILE_BYTES;
#pragma unroll
      for (int k = 0; k < B128S; ++k) {
        const size_t off = base + (size_t)(tid * 16 + k * (NTHREADS * 16));
        if (off + 16 <= totalBytes)
          async_copy_b128((const char*)target + off,
                          (char*)buf[(i + 1) & 1] + tid * 16 + k * (NTHREADS * 16));
      }
      wait_async<B128S>();
    } else {
      wait_async<0>();
    }
    __syncthreads();  // all waves' async writes for tile i are visible

    const float* ld = (const float*)buf[i & 1];
#pragma unroll
    for (int j = 0; j < PPT; ++j) {
      const int p = tid + j * NTHREADS;          // 24B stride -> conflict-free in wave32
      const int gp = tile * TILE_PAIRS + p;
      if (gp < nPairs) {
        const float2* q = (const float2*)(ld + 6 * p);  // 8B aligned
        const float2 a = q[0];   // w, label
        const float2 b = q[1];   // x1, y1
        const float2 c = q[2];   // x2, y2
        const int x1 = min(1023, max(0, (int)floorf(1024.f * b.x)));
        const int y1 = min(1023, max(0, (int)floorf(1024.f * b.y)));
        const int x2 = min(1023, max(0, (int)floorf(1024.f * c.x)));
        const int y2 = min(1023, max(0, (int)floorf(1024.f * c.y)));
        const float p1 = img[(y1 << 10) | x1];   // L2-resident gather
        const float p2 = img[(y2 << 10) | x2];
        const float ratio = p1 / (p2 + 1e-7f);
        const int predict = (ratio > 1.12f) ? 2
                          : ((ratio < 0.892857142857142857f) ? 1 : 0);
        const int lab = (int)a.y;
        accWM += (predict != lab) ? a.x : 0.f;
        accW  += a.x;
      }
    }
    __syncthreads();  // done reading buf[i&1] before it is overwritten (i+2)
  }

  // wave32 butterfly reduction (ds permute), then cross-wave via LDS
#pragma unroll
  for (int off = 16; off > 0; off >>= 1) {
    accWM += __shfl_xor(accWM, off, 32);
    accW  += __shfl_xor(accW,  off, 32);
  }
  if ((tid & 31) == 0) { wavePart[tid >> 5][0] = accWM; wavePart[tid >> 5][1] = accW; }
  __syncthreads();
  if (tid == 0) {
    float a = 0.f, b = 0.f;
#pragma unroll
    for (int w = 0; w < NTHREADS / 32; ++w) { a += wavePart[w][0]; b += wavePart[w][1]; }
    partials[2 * bid + 0] = a;
    partials[2 * bid + 1] = b;
  }
}

__global__ __launch_bounds__(1024) void whdr_finalize_kernel(
    const float* __restrict__ partials, float* __restrict__ out, int nParts) {
  __shared__ float wp[32][2];
  const int tid = threadIdx.x;
  float a = 0.f, b = 0.f;
  for (int i = tid; i < nParts; i += 1024) { a += partials[2 * i]; b += partials[2 * i + 1]; }
#pragma unroll
  for (int off = 16; off > 0; off >>= 1) {
    a += __shfl_xor(a, off, 32);
    b += __shfl_xor(b, off, 32);
  }
  if ((tid & 31) == 0) { wp[tid >> 5][0] = a; wp[tid >> 5][1] = b; }
  __syncthreads();
  if (tid == 0) {
    float A = 0.f, B = 0.f;
#pragma unroll
    for (int i = 0; i < 32; ++i) { A += wp[i][0]; B += wp[i][1]; }
    out[0] = A / B;
  }
}

extern "C" void kernel_launch(void* const* d_in, const int* in_sizes, int n_in,
                              void* d_out, int out_size, void* d_ws, size_t ws_size,
                              hipStream_t stream) {
  const float* img    = (const float*)d_in[0];   // (1,1,1024,1024) f32
  const float* target = (const float*)d_in[1];   // (1,N,6) f32
  float* out = (float*)d_out;                    // 1 f32
  float* partials = (float*)d_ws;                // 2*NBLOCKS f32 = 8 KB scratch

  const int nPairs = in_sizes[1] / 6;
  const int nTiles = (nPairs + TILE_PAIRS - 1) / TILE_PAIRS;

  whdr_reduce_kernel<<<NBLOCKS, NTHREADS, 0, stream>>>(img, target, partials,
                                                       nPairs, nTiles);
  whdr_finalize_kernel<<<1, 1024, 0, stream>>>(partials, out, NBLOCKS);
}